// NUFTIFFTPolygonEncoder_15461882265818
// MI455X (gfx1250) — compile-verified
//
#include <hip/hip_runtime.h>
#include <math.h>

typedef __attribute__((ext_vector_type(2))) float v2f;
typedef __attribute__((ext_vector_type(8))) float v8f;

#define NVERT 64
#define NKY   33
#define KYP   48            // ky padded to 3 WMMA tiles
#define NFREQ (64 * NKY)    // 2112

// One workgroup per batch: phase A (edge NUFT into LDS F), then irfft2 as two
// WMMA-f32 GEMM stages entirely in LDS.
__global__ __launch_bounds__(256) void nuft_ifft_poly_kernel(
    const float* __restrict__ poly, float* __restrict__ out)
{
    __shared__ float vx[NVERT], vy[NVERT], cd[NVERT];
    __shared__ float tab_c[64], tab_s[64], tab_ns[64];  // cos, sin, -sin
    __shared__ float wq[KYP];                           // irfft ky weights
    __shared__ float Fr[64 * KYP], Fi[64 * KYP];
    __shared__ float Gr[64 * KYP], Gi[64 * KYP];
    __shared__ float s_mass;

    const int b   = blockIdx.x;
    const int tid = threadIdx.x;

    // ---- load vertices, shifted by -EXTENT.min = (+1,+1) ----
    if (tid < NVERT) {
        vx[tid] = poly[(b * NVERT + tid) * 2 + 0] + 1.0f;
        vy[tid] = poly[(b * NVERT + tid) * 2 + 1] + 1.0f;
    }
    // twiddle tables: cos/sin(2*pi*j/64)
    if (tid < 64) {
        float s, c;
        sincosf((float)tid * 0.09817477042468103f, &s, &c);
        tab_c[tid]  = c;
        tab_s[tid]  = s;
        tab_ns[tid] = -s;
    }
    if (tid < KYP)
        wq[tid] = (tid == 0 || tid == 32) ? 1.0f : ((tid < NKY) ? 2.0f : 0.0f);
    for (int i = tid; i < 64 * KYP; i += 256) { Fr[i] = 0.0f; Fi[i] = 0.0f; }
    __syncthreads();

    if (tid < NVERT) {
        int n1 = (tid + 1) & 63;
        cd[tid] = 0.5f * (vx[tid] * vy[n1] - vy[tid] * vx[n1]);
    }
    __syncthreads();
    if (tid == 0) {
        float m = 0.0f;
        for (int n = 0; n < NVERT; ++n) m += cd[n];
        s_mass = m;
    }
    __syncthreads();

    // ---- Phase A: F[kx,ky] = sum_n CD_n * (-2) * sum_t exp(-i sig_t)/den_t ----
    const float PI = 3.14159265358979323846f;
    for (int t = tid; t < NFREQ; t += 256) {
        int   kxi = t / NKY;
        int   kyi = t - kxi * NKY;
        float kxv = (kxi <= 31) ? (float)kxi : (float)(kxi - 64); // fftfreq
        float kyv = (float)kyi;                                   // rfftfreq
        float wx = PI * kxv, wy = PI * kyv;                       // omega = pi*K

        float fr = 0.0f, fi = 0.0f;
        float ang_p = wx * vx[0] + wy * vy[0];
        float sp, cp;
        sincosf(ang_p, &sp, &cp);
        for (int n = 0; n < NVERT; ++n) {
            int   n1    = (n + 1) & 63;
            float ang_n = wx * vx[n1] + wy * vy[n1];
            float sn, cn;
            sincosf(ang_n, &sn, &cn);
            float s1 = ang_p, s2 = ang_n;           // s0 == 0
            float d0 = s1 * s2;
            float d1 = s1 * (s1 - s2);
            float d2 = s2 * (s2 - s1);
            float i0 = (fabsf(d0) > 1e-8f) ? 1.0f / d0 : 0.0f;
            float i1 = (fabsf(d1) > 1e-8f) ? 1.0f / d1 : 0.0f;
            float i2 = (fabsf(d2) > 1e-8f) ? 1.0f / d2 : 0.0f;
            float c  = cd[n];
            fr += c * (i0 + cp * i1 + cn * i2);     // Re(exp(-is)) = cos
            fi += c * (-(sp * i1) - (sn * i2));     // Im(exp(-is)) = -sin
            ang_p = ang_n; sp = sn; cp = cn;
        }
        fr *= -2.0f;
        fi *= -2.0f;
        if (t == 0) { fr = s_mass; fi = 0.0f; }     // F[0,0] = mass
        float r2   = kxv * kxv + kyv * kyv;
        float filt = expf(-r2 * (1.0f / 512.0f));   // exp(-0.5*(4|K|/64)^2)
        Fr[kxi * KYP + kyi] = fr * filt;
        Fi[kxi * KYP + kyi] = fi * filt;
    }
    __syncthreads();

    // ---- WMMA lane decomposition (wave32, f32 16x16x4 layout) ----
    // Scalarize the wave id so job loops become SALU control flow and every
    // WMMA executes with EXEC all-ones (ISA requirement).
    const int wave = __builtin_amdgcn_readfirstlane(tid >> 5);
    const int lane = tid & 31;
    const int h    = lane >> 4;   // lane half: k offset
    const int lm   = lane & 15;   // row (A) / col (B,D) within tile

    // ---- Stage 1: G[x,ky] = sum_kx (cos + i sin)(2pi kx x/64) * F[kx,ky] ----
    // 24 tile jobs: 4 m-tiles x 3 n-tiles x {Re,Im}, 3 per wave.
    // Gr = C*Fr + (-S)*Fi ; Gi = S*Fr + C*Fi -- operand tables chosen per job,
    // hoisted out of the k-loop (no per-iteration selects/negates).
    for (int j = wave; j < 24; j += 8) {
        const int out_im = (j >= 12);
        const int tile   = out_im ? (j - 12) : j;
        const int mt = tile / 3, nt = tile % 3;
        const int m  = mt * 16 + lm;
        const int nn = nt * 16 + lm;
        const float* tA1 = out_im ? tab_s : tab_c;
        const float* tA2 = out_im ? tab_c : tab_ns;
        v8f acc = {};
        for (int k0 = 0; k0 < 64; k0 += 4) {
            int ka = k0 + 2 * h;
            int kb = ka + 1;
            v2f A1 = { tA1[(m * ka) & 63], tA1[(m * kb) & 63] };
            v2f A2 = { tA2[(m * ka) & 63], tA2[(m * kb) & 63] };
            v2f bR = { Fr[ka * KYP + nn], Fr[kb * KYP + nn] };
            v2f bI = { Fi[ka * KYP + nn], Fi[kb * KYP + nn] };
            acc = __builtin_amdgcn_wmma_f32_16x16x4_f32(false, A1, false, bR,
                                                        (short)0, acc, false, false);
            acc = __builtin_amdgcn_wmma_f32_16x16x4_f32(false, A2, false, bI,
                                                        (short)0, acc, false, false);
        }
        float* Gd = out_im ? Gi : Gr;
#pragma unroll
        for (int v = 0; v < 8; ++v)
            Gd[(mt * 16 + v + 8 * h) * KYP + nn] = acc[v];
    }
    __syncthreads();

    // ---- Stage 2: f[x,y] = sum_ky w_k*(Gr*cos + Gi*(-sin))(2pi ky y/64) ----
    // 16 output tiles (4x4), 2 per wave. Final scale = 1024/4096 = 0.25.
    for (int j = wave; j < 16; j += 8) {
        const int mt = j >> 2, nt = j & 3;
        const int m  = mt * 16 + lm;
        const int nn = nt * 16 + lm;
        v8f acc = {};
        for (int k0 = 0; k0 < 48; k0 += 4) {
            int   ka  = k0 + 2 * h;
            int   kb  = ka + 1;
            float wka = wq[ka], wkb = wq[kb];
            v2f aR = { Gr[m * KYP + ka], Gr[m * KYP + kb] };
            v2f aI = { Gi[m * KYP + ka], Gi[m * KYP + kb] };
            v2f bC = { wka * tab_c[(ka * nn) & 63],  wkb * tab_c[(kb * nn) & 63] };
            v2f bS = { wka * tab_ns[(ka * nn) & 63], wkb * tab_ns[(kb * nn) & 63] };
            acc = __builtin_amdgcn_wmma_f32_16x16x4_f32(false, aR, false, bC,
                                                        (short)0, acc, false, false);
            acc = __builtin_amdgcn_wmma_f32_16x16x4_f32(false, aI, false, bS,
                                                        (short)0, acc, false, false);
        }
#pragma unroll
        for (int v = 0; v < 8; ++v)
            out[b * 4096 + (mt * 16 + v + 8 * h) * 64 + nn] = 0.25f * acc[v];
    }
}

extern "C" void kernel_launch(void* const* d_in, const int* in_sizes, int n_in,
                              void* d_out, int out_size, void* d_ws, size_t ws_size,
                              hipStream_t stream) {
    (void)n_in; (void)out_size; (void)d_ws; (void)ws_size;
    const float* poly = (const float*)d_in[0];
    float*       out  = (float*)d_out;
    int B = in_sizes[0] / (NVERT * 2);   // 128
    nuft_ifft_poly_kernel<<<dim3(B), dim3(256), 0, stream>>>(poly, out);
}